// GlobalConv_61418032332910
// MI455X (gfx1250) — compile-verified
//
#include <hip/hip_runtime.h>

typedef __attribute__((ext_vector_type(2))) float v2f;
typedef __attribute__((ext_vector_type(8))) float v8f;

namespace {
constexpr int kB = 4;
constexpr int kN = 128;
constexpr int kHeads = 16;
constexpr int kHid = 32;
constexpr int kWN = 2304;   // W_NUMEL = 1024+512+256+512
constexpr int kDimIn = 80;
constexpr int kNodes = 4;   // nodes per workgroup -> 16 fully used G rows
constexpr int kThreads = 256;  // 8 waves

constexpr float kW1Scale  = 0.25f;                  // 1/sqrt(HEADS=16)
constexpr float kW2Scale  = 0.17677669529663687f;   // 1/sqrt(HID=32)
constexpr float kInvSqrt3 = 0.5773502691896258f;
constexpr float kC0       = 0.14433756729740643f;   // sqrt(1/48)
constexpr float kC1       = 0.25f;                  // sqrt(3/48)
constexpr float kC0s = kC0 * kW2Scale;              // W2 scale folded out of stage 3
constexpr float kC1s = kC1 * kInvSqrt3 * kW2Scale;

__device__ inline v8f wmma4(v2f a, v2f b, v8f c) {
  // D = A(16x4,f32) * B(4x16,f32) + C(16x16,f32), exact fp32 math.
  return __builtin_amdgcn_wmma_f32_16x16x4_f32(false, a, false, b, (short)0, c,
                                               false, false);
}
}  // namespace

__global__ __launch_bounds__(kThreads)
void global_conv_kernel(const float* __restrict__ nodef,
                        const float* __restrict__ edge,
                        const float* __restrict__ atten,
                        const float* __restrict__ nnorm,
                        const float* __restrict__ W1,
                        const float* __restrict__ W2,
                        float* __restrict__ out) {
  // Phase buffers (stages 0-2) overlapped with Weff (stages 3-4).
  __shared__ union SMem {
    struct {
      float A[kNodes * kN * kHeads];   // per-node 128x16 atten slice
      float W1s[kHeads * kHid];        // 16x32 scaled W1
      float H[kNodes * kN * kHid];     // per-node 128x32 relu(A @ W1s)
      float S[kNodes * 4 * kN];        // per-node 4x128 scalar channels
    } p;
    float Weff[16 * kWN];              // row = node*4 + channel
  } sm;
  __shared__ float sG[16 * kHid];      // row = node*4 + channel (all 16 used)
  __shared__ float sX[kNodes * kDimIn];

  const int tid  = threadIdx.x;
  const int lane = tid & 31;
  const int wv   = tid >> 5;          // 0..7
  const int l15  = lane & 15;
  const int hi   = lane >> 4;         // half-wave: K offset +2 / M offset +8
  const int n0   = blockIdx.x * kNodes;
  const int b    = blockIdx.y;

  // ---------------- stage 0: cooperative loads into LDS ----------------
  for (int i = tid; i < kHeads * kHid; i += kThreads) sm.p.W1s[i] = W1[i] * kW1Scale;
  for (int i = tid; i < kNodes * kDimIn; i += kThreads) {
    const int ni = i / kDimIn, d = i - ni * kDimIn;
    sX[i] = nodef[(b * kN + n0 + ni) * kDimIn + d];
  }
  for (int i = tid; i < kNodes * kN; i += kThreads) {
    const int ni = i >> 7, m = i & (kN - 1);
    // edge_attr[b, m, n0+ni, 0:4]: 16B contiguous
    const float4 e = *(const float4*)(edge + (((b * kN + m) * kN + n0 + ni) << 2));
    float* S = sm.p.S + ni * 4 * kN;
    S[0 * kN + m] = e.x; S[1 * kN + m] = e.y;
    S[2 * kN + m] = e.z; S[3 * kN + m] = e.w;
    // atten[b, m, n0+ni, 0:16]: 64B contiguous
    const float4* ap = (const float4*)(atten + (((b * kN + m) * kN + n0 + ni) << 4));
    float4* dp = (float4*)(sm.p.A + (ni * kN + m) * kHeads);
    dp[0] = ap[0]; dp[1] = ap[1]; dp[2] = ap[2]; dp[3] = ap[3];
  }
  __syncthreads();

  // ---------------- stage 1: H = relu(A(128x16) @ W1s(16x32)) per node -----
  // wave -> (node = wv>>1, n-tile = wv&1), covers all 8 m-tiles.
  {
    const int ni = wv >> 1, nt = wv & 1;
    const float* A = sm.p.A + ni * kN * kHeads;
    float* H = sm.p.H + ni * kN * kHid;
    v2f bw[4];  // loop-invariant B fragments
#pragma unroll
    for (int kk = 0; kk < 4; ++kk) {
      const int kb = kk * 4 + hi * 2;
      bw[kk].x = sm.p.W1s[kb * kHid + nt * 16 + l15];
      bw[kk].y = sm.p.W1s[(kb + 1) * kHid + nt * 16 + l15];
    }
    for (int mt = 0; mt < 8; ++mt) {
      v8f acc = {};
#pragma unroll
      for (int kk = 0; kk < 4; ++kk) {
        const int kb = kk * 4 + hi * 2;
        v2f a;
        a.x = A[(mt * 16 + l15) * kHeads + kb];
        a.y = A[(mt * 16 + l15) * kHeads + kb + 1];
        acc = wmma4(a, bw[kk], acc);
      }
#pragma unroll
      for (int v = 0; v < 8; ++v) {
        const float x = acc[v];
        H[(mt * 16 + hi * 8 + v) * kHid + nt * 16 + l15] = x > 0.f ? x : 0.f;
      }
    }
  }
  __syncthreads();

  // ---------------- stage 2: G rows = S_pad(16x128) @ H(128x32) per node ---
  {
    const int ni = wv >> 1, nt = wv & 1;
    const float* S = sm.p.S + ni * 4 * kN;
    const float* H = sm.p.H + ni * kN * kHid;
    v8f acc = {};
    for (int kk = 0; kk < 32; ++kk) {
      const int m = kk * 4 + hi * 2;
      v2f a, bb;
      const bool valid = (l15 < 4);
      a.x = valid ? S[l15 * kN + m] : 0.f;
      a.y = valid ? S[l15 * kN + m + 1] : 0.f;
      bb.x = H[m * kHid + nt * 16 + l15];
      bb.y = H[(m + 1) * kHid + nt * 16 + l15];
      acc = wmma4(a, bb, acc);
    }
    if (hi == 0) {  // rows 0..3 = the node's 4 channels
#pragma unroll
      for (int v = 0; v < 4; ++v)
        sG[(ni * 4 + v) * kHid + nt * 16 + l15] = acc[v];
    }
  }
  __syncthreads();

  // ---------------- stage 3: Weff(16x2304) = sG(16x32) @ W2(32x2304) -------
  // All 16 rows meaningful now (4 nodes x 4 channels). 144 col-tiles / 8 waves.
  for (int t = wv; t < kWN / 16; t += 8) {
    const int col = t * 16 + l15;
    v8f acc = {};
#pragma unroll
    for (int kk = 0; kk < 8; ++kk) {
      const int kb = kk * 4 + hi * 2;
      v2f a, bb;
      a.x = sG[l15 * kHid + kb];
      a.y = sG[l15 * kHid + kb + 1];
      bb.x = W2[kb * kWN + col];
      bb.y = W2[(kb + 1) * kWN + col];
      acc = wmma4(a, bb, acc);
    }
#pragma unroll
    for (int v = 0; v < 8; ++v)
      sm.Weff[(hi * 8 + v) * kWN + col] = acc[v];
  }
  __syncthreads();

  // ---------------- stage 4: per-node outputs (4 x 80 values) --------------
  for (int i = tid; i < kNodes * kDimIn; i += kThreads) {
    const int ni = i / kDimIn, o = i - ni * kDimIn;
    const float invM = 1.0f / nnorm[0];
    const float* x0 = sX + ni * kDimIn;  // 32
    const float* x1 = x0 + 32;           // [i*3+a], 16x3
    const float* Wn = sm.Weff + (ni * 4) * kWN;  // this node's 4 channel rows
    float res;
    if (o < 32) {
      const int j = o;
      float acc = 0.f, acc2 = 0.f;
      for (int i2 = 0; i2 < 32; ++i2)           // w00 (s0 channel)
        acc += x0[i2] * Wn[0 * kWN + i2 * 32 + j];
      for (int i2 = 0; i2 < 16; ++i2)           // w11 (s1 channels)
        for (int a = 0; a < 3; ++a)
          acc2 += x1[i2 * 3 + a] * Wn[(1 + a) * kWN + 1792 + i2 * 32 + j];
      res = kC0s * (acc + kInvSqrt3 * acc2);
    } else {
      const int j = (o - 32) / 3, a = (o - 32) % 3;
      float acc = 0.f;
      for (int i2 = 0; i2 < 32; ++i2)           // w01 (s1 channels)
        acc += x0[i2] * Wn[(1 + a) * kWN + 1024 + i2 * 16 + j];
      for (int i2 = 0; i2 < 16; ++i2)           // w10 (s0 channel)
        acc += x1[i2 * 3 + a] * Wn[0 * kWN + 1536 + i2 * 16 + j];
      res = kC1s * acc;
    }
    out[(b * kN + n0 + ni) * kDimIn + o] = res * invM;
  }
}

extern "C" void kernel_launch(void* const* d_in, const int* in_sizes, int n_in,
                              void* d_out, int out_size, void* d_ws, size_t ws_size,
                              hipStream_t stream) {
  const float* nodef = (const float*)d_in[0];
  const float* edge  = (const float*)d_in[1];
  const float* atten = (const float*)d_in[2];
  const float* nnorm = (const float*)d_in[3];
  const float* W1    = (const float*)d_in[4];
  const float* W2    = (const float*)d_in[5];
  float* out = (float*)d_out;
  dim3 grid(kN / kNodes, kB);
  global_conv_kernel<<<grid, kThreads, 0, stream>>>(nodef, edge, atten, nnorm, W1, W2, out);
  (void)in_sizes; (void)n_in; (void)out_size; (void)d_ws; (void)ws_size;
}